// Dice_Loss_by_block_89532888252943
// MI455X (gfx1250) — compile-verified
//
#include <hip/hip_runtime.h>
#include <hip/hip_bf16.h>

// ---------------------------------------------------------------------------
// Dice loss by block, MI455X (gfx1250).
// Memory-bound segmented reduction: 201 MB @ 23.3 TB/s => ~8.6 us floor.
// CDNA5 paths used:
//   * global_load_async_to_lds_b128 + s_wait_asynccnt (double-buffered stage)
//   * conflict-free per-lane LDS ds_add_f32 atomics into replicated bins
//   * V_WMMA_F32_16X16X4_F32 ones-matrix reduction of the 64 bin copies
// ---------------------------------------------------------------------------

#define TPB          256
#define WPB          256                      // workgroups per batch
#define ITERS        8                        // float4 iterations per thread
#define F4_PER_BATCH (128 * 128 * 128 / 4)    // 524288 float4 per batch

#define NSEG   16                             // padded segment count (0..10 used)
#define NPAIR  48                             // 3 quantities * 16 segments
#define NCOPY  64                             // lane-private bin copies

#define LDS_BINS_BYTES (NPAIR * NCOPY * 4)    // 12288
#define LDS_X    (LDS_BINS_BYTES)             // 12288  (2 bufs * 256 thr * 16B)
#define LDS_T    (LDS_X + 2 * TPB * 16)       // 20480
#define LDS_S    (LDS_T + 2 * TPB * 16)       // 28672
#define LDS_PRES (LDS_S + 2 * TPB * 16)       // 36864
#define LDS_TOTAL (LDS_PRES + 16)             // 36880 bytes dynamic LDS

#define DICE_EPS 1e-6f

typedef float v2f __attribute__((ext_vector_type(2)));
typedef float v8f __attribute__((ext_vector_type(8)));

// CDNA5 async global->LDS copy (ASYNCcnt tracked). GV addressing mode:
// LDS address is the LDS base plus the vdst VGPR; global addr in VGPR pair.
__device__ __forceinline__ void async_load_b128(unsigned lds_off, const void* gaddr) {
    asm volatile("global_load_async_to_lds_b128 %0, %1, off"
                 :: "v"(lds_off), "v"((unsigned long long)(size_t)gaddr)
                 : "memory");
}

__global__ void zero_ws_kernel(int* ws) {
    int i = blockIdx.x * blockDim.x + threadIdx.x;
    if (i < 8 * NPAIR + 1) ws[i] = 0;   // 384 float bins + 1 presence mask
}

__global__ __launch_bounds__(TPB)
void dice_accum_kernel(const float* __restrict__ x,
                       const float* __restrict__ t,
                       const int*   __restrict__ s,
                       float* __restrict__ gbins,   // [8][48]
                       int*   __restrict__ gmask) {
    extern __shared__ char smem[];
    const unsigned tid   = threadIdx.x;
    const unsigned batch = blockIdx.y;
    const unsigned wg    = blockIdx.x;

    float* binsf = (float*)smem;
    const unsigned lds_base = (unsigned)(size_t)smem;   // LDS byte offset of dyn-LDS
    const unsigned cid = tid & (NCOPY - 1);             // lane-private copy index

    // Zero the replicated bins (768 float4 / 256 threads = 3 each) + presence.
    float4* b4 = (float4*)smem;
    #pragma unroll
    for (int k = 0; k < LDS_BINS_BYTES / 16 / TPB; ++k)
        b4[tid + k * TPB] = make_float4(0.f, 0.f, 0.f, 0.f);
    if (tid == 0) *(int*)(smem + LDS_PRES) = 0;
    __syncthreads();

    // This workgroup's contiguous chunk (float4 units) inside its batch.
    const size_t f4base = (size_t)batch * F4_PER_BATCH + (size_t)wg * (TPB * ITERS);
    const char* px = (const char*)x + f4base * 16;
    const char* pt = (const char*)t + f4base * 16;
    const char* ps = (const char*)s + f4base * 16;

    auto issue = [&](int i) {
        const unsigned goff = (unsigned)(i * TPB + tid) * 16u;
        const unsigned loff = (((unsigned)(i & 1)) * TPB + tid) * 16u;
        async_load_b128(lds_base + LDS_X + loff, px + goff);
        async_load_b128(lds_base + LDS_T + loff, pt + goff);
        async_load_b128(lds_base + LDS_S + loff, ps + goff);
    };

    unsigned pres = 0u;
    issue(0);
    for (int i = 0; i < ITERS; ++i) {
        if (i + 1 < ITERS) {
            issue(i + 1);
            // 6 async ops in flight; <=3 means current buffer's 3 are done.
            asm volatile("s_wait_asynccnt 3" ::: "memory");
        } else {
            asm volatile("s_wait_asynccnt 0" ::: "memory");
        }
        const unsigned loff = (((unsigned)(i & 1)) * TPB + tid) * 16u;
        const float4 xv = *(const float4*)(smem + LDS_X + loff);
        const float4 tv = *(const float4*)(smem + LDS_T + loff);
        const int4   sv = *(const int4*)  (smem + LDS_S + loff);

        #define ACC_ELEM(XX, TT, SS)                                            \
            {                                                                   \
                const unsigned sg = (unsigned)(SS) & 15u;                       \
                pres |= (1u << sg);                                             \
                atomicAdd(&binsf[(0u * NSEG + sg) * NCOPY + cid], (XX) * (TT)); \
                atomicAdd(&binsf[(1u * NSEG + sg) * NCOPY + cid], (XX));        \
                atomicAdd(&binsf[(2u * NSEG + sg) * NCOPY + cid], (TT));        \
            }
        ACC_ELEM(xv.x, tv.x, sv.x)
        ACC_ELEM(xv.y, tv.y, sv.y)
        ACC_ELEM(xv.z, tv.z, sv.z)
        ACC_ELEM(xv.w, tv.w, sv.w)
        #undef ACC_ELEM
    }

    atomicOr((int*)(smem + LDS_PRES), (int)pres);
    __syncthreads();

    if (tid == 0) atomicOr(gmask, *(const int*)(smem + LDS_PRES));

    // Wave 0 reduces the 64 copies of each of 48 bins.
    if (tid < 32) {
        const unsigned lane = tid;
#if __has_builtin(__builtin_amdgcn_wmma_f32_16x16x4_f32)
        // Ones-matrix WMMA reduction: A[m][k] = partials (pair m, copies 4i+..),
        // B = ones  =>  D[m][n] = sum_k A[m][k], exact f32.
        // A 16x4 f32 layout: lane m holds K={0,1}; lane m+16 holds K={2,3}.
        v2f ones; ones[0] = 1.0f; ones[1] = 1.0f;
        for (int g = 0; g < 3; ++g) {
            v8f acc = {0.f, 0.f, 0.f, 0.f, 0.f, 0.f, 0.f, 0.f};
            const float* base =
                binsf + (g * 16 + (lane & 15)) * NCOPY + 2 * (lane >> 4);
            #pragma unroll
            for (int i = 0; i < NCOPY / 4; ++i) {
                const float2 d = *(const float2*)(base + i * 4);
                v2f a; a[0] = d.x; a[1] = d.y;
                acc = __builtin_amdgcn_wmma_f32_16x16x4_f32(
                    false, a, false, ones, (short)0, acc, false, false);
            }
            // D layout: vgpr r, lanes 0-15 -> M=r ; lanes 16-31 -> M=r+8.
            if (lane < 8)
                atomicAdd(&gbins[batch * NPAIR + g * 16 + lane], acc[lane]);
            else if (lane >= 16 && lane < 24)
                atomicAdd(&gbins[batch * NPAIR + g * 16 + 8 + (lane - 16)],
                          acc[lane - 16]);
        }
#else
        for (unsigned p = lane; p < NPAIR; p += 32) {
            float sum = 0.f;
            for (int c = 0; c < NCOPY; ++c) sum += binsf[p * NCOPY + c];
            atomicAdd(&gbins[batch * NPAIR + p], sum);
        }
#endif
    }
}

__global__ void dice_finalize_kernel(const float* __restrict__ gbins,
                                     const int* __restrict__ gmask,
                                     float* __restrict__ out, int out_size) {
    if (threadIdx.x == 0 && blockIdx.x == 0) {
        const int mask = *gmask;
        float total = 0.f, num = 0.f;
        for (int k = 1; k <= 10; ++k) {
            float sum_loss = 0.f, valid = 0.f;
            for (int b = 0; b < 8; ++b) {
                const float I = gbins[b * NPAIR + 0 * NSEG + k];
                const float X = gbins[b * NPAIR + 1 * NSEG + k];
                const float T = gbins[b * NPAIR + 2 * NSEG + k];
                float bl = 1.0f - 2.0f * I / (X + T + 2.0f * DICE_EPS);
                bl = (T == 0.0f) ? 0.0f : bl;
                valid += (T != 0.0f) ? 1.0f : 0.0f;
                sum_loss += bl;
            }
            const float lpb = sum_loss / fmaxf(valid, 1.0f);
            const bool present = (mask >> k) & 1;
            total += present ? lpb : 0.0f;
            num += present ? 1.0f : 0.0f;
        }
        out[0] = total / num;
    }
    const int idx = blockIdx.x * blockDim.x + threadIdx.x;
    if (idx >= 1 && idx < out_size) out[idx] = 0.0f;   // second tuple element
}

extern "C" void kernel_launch(void* const* d_in, const int* in_sizes, int n_in,
                              void* d_out, int out_size, void* d_ws, size_t ws_size,
                              hipStream_t stream) {
    const float* x = (const float*)d_in[0];
    const float* t = (const float*)d_in[1];
    const int*   s = (const int*)d_in[2];

    float* gbins = (float*)d_ws;             // [8][48] floats
    int*   gmask = ((int*)d_ws) + 8 * NPAIR; // presence bitmask

    zero_ws_kernel<<<2, 256, 0, stream>>>((int*)d_ws);

    dim3 grid(WPB, 8, 1);
    dice_accum_kernel<<<grid, TPB, LDS_TOTAL, stream>>>(x, t, s, gbins, gmask);

    dice_finalize_kernel<<<1, 64, 0, stream>>>(gbins, gmask, (float*)d_out, out_size);
}